// VCGAuctioneer_59450937311878
// MI455X (gfx1250) — compile-verified
//
#include <hip/hip_runtime.h>
#include <cstdint>

#define NUM_EXPERTS 64
#define TPB 256           // tokens per block == threads per block (8 wave32s)
#define ROW_STRIDE 66     // floats: 264B rows -> 8B-aligned, LDS-bank-conflict-free reads

#if __has_builtin(__builtin_amdgcn_global_load_async_to_lds_b64)
#define HAVE_ASYNC 1
#else
#define HAVE_ASYNC 0
#endif

// Async-LDS DMA builtin wants pointers to an 8-byte int vector type.
typedef int v2i __attribute__((vector_size(2 * sizeof(int))));
typedef __attribute__((address_space(1))) v2i* g_v2i_ptr;   // global
typedef __attribute__((address_space(3))) v2i* l_v2i_ptr;   // LDS

__global__ __launch_bounds__(TPB)
void vcg_route_kernel(const float* __restrict__ conf,
                      const float* __restrict__ wealth,
                      int*   __restrict__ out_idx,
                      float* __restrict__ out_w,
                      float* __restrict__ out_pay,
                      int n_tokens)
{
    __shared__ float tile[TPB * ROW_STRIDE];
    __shared__ float wl[NUM_EXPERTS];

    const int tid  = threadIdx.x;
    const int lane = tid & 31;
    const int wv   = tid >> 5;
    const long long tok0 = (long long)blockIdx.x * TPB;

    if (tid < NUM_EXPERTS) wl[tid] = wealth[tid];

    // ---- Stage TPB rows x 64 floats into LDS.
    // Wave w loads rows {w, w+8, ...}; lane covers one 8-byte chunk -> each
    // async op moves one full 256B row per wave, fully coalesced.
#pragma unroll 4
    for (int it = 0; it < TPB / 8; ++it) {
        const int row = wv + it * 8;
        const long long tok = tok0 + row;
        if (tok < (long long)n_tokens) {
            const float* gsrc = conf + tok * NUM_EXPERTS + lane * 2;
            float*       ldst = &tile[row * ROW_STRIDE + lane * 2];
#if HAVE_ASYNC
            __builtin_amdgcn_global_load_async_to_lds_b64(
                (g_v2i_ptr)gsrc, (l_v2i_ptr)ldst, /*offset=*/0, /*cpol=*/0);
#else
            *(float2*)ldst = *(const float2*)gsrc;
#endif
        }
    }
#if HAVE_ASYNC
#if __has_builtin(__builtin_amdgcn_s_wait_asynccnt)
    __builtin_amdgcn_s_wait_asynccnt(0);
#else
    asm volatile("s_wait_asynccnt 0" ::: "memory");
#endif
#endif
    __syncthreads();

    const long long tok = tok0 + tid;
    if (tok >= (long long)n_tokens) return;

    const float* myrow = &tile[tid * ROW_STRIDE];

    // ---- Pass 1: top-3 bids (values), top-2 indices. Strict '>' keeps
    // earliest index on ties, matching jax.lax.top_k.
    float t0 = -__builtin_inff(), t1 = -__builtin_inff(), t2 = -__builtin_inff();
    int   i0 = 0, i1 = 0;

#define UPD(bv, ev)                                                    \
    do {                                                               \
        float _b = (bv);                                               \
        if (_b > t2) {                                                 \
            if (_b > t0)      { t2 = t1; t1 = t0; i1 = i0; t0 = _b; i0 = (ev); } \
            else if (_b > t1) { t2 = t1; t1 = _b; i1 = (ev); }         \
            else              { t2 = _b; }                             \
        }                                                              \
    } while (0)

#pragma unroll
    for (int e = 0; e < NUM_EXPERTS; e += 2) {
        const float2 c = *(const float2*)&myrow[e];
        const float2 w = *(const float2*)&wl[e];   // LDS broadcast (same addr all lanes)
        UPD(c.x * w.x, e);
        UPD(c.y * w.y, e + 1);
    }
#undef UPD

    // ---- Pass 2: softmax denominator (numerically-stable, max = t0).
    float z = 0.0f;
#pragma unroll
    for (int e = 0; e < NUM_EXPERTS; e += 2) {
        const float2 c = *(const float2*)&myrow[e];
        const float2 w = *(const float2*)&wl[e];
        z += __expf(c.x * w.x - t0) + __expf(c.y * w.y - t0);
    }

    const float invz = 1.0f / z;
    const float p0 = invz;                       // exp(t0 - t0) / z
    const float p1 = __expf(t1 - t0) * invz;
    const float inv_den = 1.0f / (p0 + p1 + 1e-8f);

    // ---- Outputs: [indices int32 | routing weights f32 | payments f32]
    const long long o = 2 * tok;
    out_idx[o]     = i0;
    out_idx[o + 1] = i1;
    out_w[o]       = p0 * inv_den;
    out_w[o + 1]   = p1 * inv_den;
    out_pay[o]     = t2;
    out_pay[o + 1] = t2;
}

extern "C" void kernel_launch(void* const* d_in, const int* in_sizes, int n_in,
                              void* d_out, int out_size, void* d_ws, size_t ws_size,
                              hipStream_t stream) {
    const float* conf   = (const float*)d_in[0];   // (B, S, E) f32
    const float* wealth = (const float*)d_in[1];   // (E,) f32

    const int n_tokens = in_sizes[0] / NUM_EXPERTS;   // B*S = 262144

    int*   out_idx = (int*)d_out;                        // (B,S,2) int32
    float* out_w   = (float*)d_out + 2LL * n_tokens;     // (B,S,2) f32
    float* out_pay = (float*)d_out + 4LL * n_tokens;     // (B,S,2) f32

    const int blocks = (n_tokens + TPB - 1) / TPB;
    vcg_route_kernel<<<blocks, TPB, 0, stream>>>(conf, wealth,
                                                 out_idx, out_w, out_pay,
                                                 n_tokens);
}